// DotAttention_31404800868669
// MI455X (gfx1250) — compile-verified
//
#include <hip/hip_runtime.h>

// Problem constants (reference: B=64, S=8192, H=512, fp32)
#define B_ 64
#define S_ 8192
#define H_ 512
#define NCHUNK 16              // S-chunks for the context partial reduction
#define CHUNK_S (S_ / NCHUNK)  // 512 rows per chunk

typedef __attribute__((ext_vector_type(2))) float v2f;
typedef __attribute__((ext_vector_type(8))) float v8f;

// ---------------------------------------------------------------------------
// Kernel 1: scores[b,s] = dot(E[b,s,:], h[b,:]), masked to -inf.
// One wave computes a 16-row tile of S via V_WMMA_F32_16X16X4_F32 (fp32 WMMA,
// no precision loss). B-matrix = h chunk broadcast over all 16 N columns.
// 512 threads = 16 waves -> 256 S-rows per workgroup; grid = B * S/256.
// ---------------------------------------------------------------------------
__global__ __launch_bounds__(512) void scores_kernel(
    const float* __restrict__ hvec,          // [B,H]
    const float* __restrict__ enc,           // [B,S,H]
    const unsigned char* __restrict__ mask,  // [B,S] (jnp.bool_ = 1 byte)
    float* __restrict__ scores)              // [B,S] -> weights region of d_out
{
    __shared__ float sh[H_];
    const int tile = blockIdx.x & 31;  // 32 tiles of 256 rows per batch
    const int b    = blockIdx.x >> 5;

    // Stage h[b] in LDS (2 KB), broadcast-read by all waves.
    if (threadIdx.x < H_) sh[threadIdx.x] = hvec[(size_t)b * H_ + threadIdx.x];
    __syncthreads();

    const int wave = threadIdx.x >> 5;
    const int lane = threadIdx.x & 31;
    const int row  = lane & 15;            // M index
    const int koff = (lane >> 4) << 1;     // lanes 0-15: K=0,1  lanes 16-31: K=2,3
    const int s0   = tile * 256 + wave * 16;

    const float* rowp = enc + ((size_t)b * S_ + (size_t)(s0 + row)) * H_;

    v8f c0 = {};
    v8f c1 = {};
#pragma unroll 2
    for (int k = 0; k < H_; k += 8) {
        // A fragments: per-lane aligned b64 loads; both half-waves together
        // consume 16 contiguous bytes of each row per step.
        v2f a0 = *(const v2f*)(rowp + k + koff);
        v2f a1 = *(const v2f*)(rowp + k + 4 + koff);
        // B fragments: h chunk broadcast across N (LDS broadcast reads).
        v2f b0; b0.x = sh[k + koff];     b0.y = sh[k + koff + 1];
        v2f b1; b1.x = sh[k + 4 + koff]; b1.y = sh[k + 4 + koff + 1];
        // Two independent accumulators break the WMMA->WMMA RAW chain.
        c0 = __builtin_amdgcn_wmma_f32_16x16x4_f32(false, a0, false, b0,
                                                   (short)0, c0, false, false);
        c1 = __builtin_amdgcn_wmma_f32_16x16x4_f32(false, a1, false, b1,
                                                   (short)0, c1, false, false);
    }

    // D layout: VGPR r -> lanes 0-15: M=r ; lanes 16-31: M=8+r. All N columns
    // identical (B was broadcast), so lane 0 holds rows 0..7 and lane 16 rows
    // 8..15 in components 0..7.
    if (lane == 0 || lane == 16) {
        const int base = s0 + (lane == 16 ? 8 : 0);
        const unsigned char* mp = mask + (size_t)b * S_ + base;
        float* op = scores + (size_t)b * S_ + base;
#pragma unroll
        for (int r = 0; r < 8; ++r) {
            float v = c0[r] + c1[r];
            op[r] = mp[r] ? v : -__builtin_inff();
        }
    }
}

// ---------------------------------------------------------------------------
// Kernel 2: in-place softmax over each row of [B,S]. One workgroup per batch,
// 256 threads x 32 elements kept in registers. Deterministic reductions
// (wave32 shuffles + LDS across the 8 waves).
// ---------------------------------------------------------------------------
__global__ __launch_bounds__(256) void softmax_kernel(float* __restrict__ w)
{
    const int b = blockIdx.x;
    const int t = threadIdx.x;
    float* p = w + (size_t)b * S_;

    float v[32];
#pragma unroll
    for (int i = 0; i < 32; ++i) v[i] = p[t + i * 256];

    float m = v[0];
#pragma unroll
    for (int i = 1; i < 32; ++i) m = fmaxf(m, v[i]);
#pragma unroll
    for (int off = 16; off > 0; off >>= 1) m = fmaxf(m, __shfl_xor(m, off, 32));

    __shared__ float redmax[8];
    __shared__ float redsum[8];
    const int wv = t >> 5, ln = t & 31;
    if (ln == 0) redmax[wv] = m;
    __syncthreads();
    if (wv == 0) {
        float x = redmax[ln & 7];
#pragma unroll
        for (int off = 4; off > 0; off >>= 1) x = fmaxf(x, __shfl_xor(x, off, 32));
        if (ln == 0) redmax[0] = x;
    }
    __syncthreads();
    m = redmax[0];

    float s = 0.f;
#pragma unroll
    for (int i = 0; i < 32; ++i) { v[i] = expf(v[i] - m); s += v[i]; }
#pragma unroll
    for (int off = 16; off > 0; off >>= 1) s += __shfl_xor(s, off, 32);
    if (ln == 0) redsum[wv] = s;
    __syncthreads();
    if (wv == 0) {
        float x = redsum[ln & 7];
#pragma unroll
        for (int off = 4; off > 0; off >>= 1) x += __shfl_xor(x, off, 32);
        if (ln == 0) redsum[0] = x;
    }
    __syncthreads();

    const float inv = 1.f / redsum[0];
#pragma unroll
    for (int i = 0; i < 32; ++i) p[t + i * 256] = v[i] * inv;
}

// ---------------------------------------------------------------------------
// Kernel 3: partial context over an S-chunk. Thread t owns H-column t, so the
// 512-thread block reads each 2 KB row of E as fully-coalesced b32 loads —
// the bandwidth-optimal shape for this pure streaming reduction.
// Grid = B * NCHUNK.
// ---------------------------------------------------------------------------
__global__ __launch_bounds__(512) void context_partial_kernel(
    const float* __restrict__ enc,       // [B,S,H]
    const float* __restrict__ w,         // [B,S] softmaxed weights
    float* __restrict__ partials)        // [B,NCHUNK,H]
{
    const int chunk = blockIdx.x & (NCHUNK - 1);
    const int b     = blockIdx.x >> 4;
    const int t     = threadIdx.x;

    __shared__ float sw[CHUNK_S];
    sw[t] = w[(size_t)b * S_ + (size_t)chunk * CHUNK_S + t];
    __syncthreads();

    const float* ep = enc + ((size_t)b * S_ + (size_t)chunk * CHUNK_S) * H_ + t;
    float acc = 0.f;
#pragma unroll 8
    for (int s = 0; s < CHUNK_S; ++s)
        acc = fmaf(sw[s], ep[(size_t)s * H_], acc);

    partials[((size_t)b * NCHUNK + chunk) * H_ + t] = acc;
}

// ---------------------------------------------------------------------------
// Kernel 4: fixed-order reduction of the partials -> context (deterministic,
// no float atomics).
// ---------------------------------------------------------------------------
__global__ __launch_bounds__(512) void context_reduce_kernel(
    const float* __restrict__ partials,  // [B,NCHUNK,H]
    float* __restrict__ ctx)             // [B,H]
{
    const int b = blockIdx.x;
    const int t = threadIdx.x;
    float s = 0.f;
#pragma unroll
    for (int c = 0; c < NCHUNK; ++c)
        s += partials[((size_t)b * NCHUNK + c) * H_ + t];
    ctx[(size_t)b * H_ + t] = s;
}

// ---------------------------------------------------------------------------
extern "C" void kernel_launch(void* const* d_in, const int* in_sizes, int n_in,
                              void* d_out, int out_size, void* d_ws, size_t ws_size,
                              hipStream_t stream)
{
    (void)in_sizes; (void)n_in; (void)out_size; (void)ws_size;

    const float* hvec          = (const float*)d_in[0];          // [B,H]
    const float* enc           = (const float*)d_in[1];          // [B,S,H]
    const unsigned char* mask  = (const unsigned char*)d_in[2];  // [B,S] bool

    float* ctx = (float*)d_out;                        // [B,H]
    float* wts = (float*)d_out + (size_t)B_ * H_;      // [B,S]
    float* partials = (float*)d_ws;                    // [B,NCHUNK,H] = 2 MB

    scores_kernel<<<B_ * (S_ / 256), 512, 0, stream>>>(hvec, enc, mask, wts);
    softmax_kernel<<<B_, 256, 0, stream>>>(wts);
    context_partial_kernel<<<B_ * NCHUNK, 512, 0, stream>>>(enc, wts, partials);
    context_reduce_kernel<<<B_, 512, 0, stream>>>(partials, ctx);
}